// SinkhornKnoppLayer_83519934038507
// MI455X (gfx1250) — compile-verified
//
#include <hip/hip_runtime.h>
#include <hip/hip_bf16.h>

// ---------------- problem constants ----------------
#define BATCH 16
#define NN    1024
#define DD    256
#define NBLK  256              // persistent kernel blocks (16 batches x 16 bands)
#define BAND  64               // rows per band
#define NITER 100

typedef __attribute__((ext_vector_type(2)))  float    v2f;
typedef __attribute__((ext_vector_type(8)))  float    v8f;
typedef __attribute__((ext_vector_type(4)))  unsigned v4u;
typedef __attribute__((ext_vector_type(4)))  int      v4i;
typedef __attribute__((ext_vector_type(8)))  int      v8i;

// device-scope (L2) load, bypassing the WGP read cache
__device__ __forceinline__ float ld_dev(const float* p) {
    return __hip_atomic_load(p, __ATOMIC_RELAXED, __HIP_MEMORY_SCOPE_AGENT);
}

// ---------------- init scratch ----------------
__global__ __launch_bounds__(256) void init_kernel(float* cbuf, unsigned* bar) {
    int i = blockIdx.x * 256 + threadIdx.x;
    if (i < 3 * BATCH * NN) cbuf[i] = 0.0f;
    if (i < 2) bar[i] = 0u;
}

// ---------------- inverse L2 norms (one wave per row) ----------------
__global__ __launch_bounds__(256) void l2norm_kernel(const float* __restrict__ fA,
                                                     const float* __restrict__ fB,
                                                     float* __restrict__ invA,
                                                     float* __restrict__ invB) {
    int gw   = (blockIdx.x * 256 + threadIdx.x) >> 5;   // global wave id = row id
    int lane = threadIdx.x & 31;
    const float* src = (gw < BATCH * NN) ? fA : fB;
    float*       dst = (gw < BATCH * NN) ? invA : invB;
    int row = gw & (BATCH * NN - 1);
    const float4* p = (const float4*)(src + (size_t)row * DD);
    float4 x0 = p[lane];
    float4 x1 = p[lane + 32];
    float s = x0.x*x0.x + x0.y*x0.y + x0.z*x0.z + x0.w*x0.w
            + x1.x*x1.x + x1.y*x1.y + x1.z*x1.z + x1.w*x1.w;
    #pragma unroll
    for (int off = 16; off > 0; off >>= 1) s += __shfl_xor(s, off, 32);
    if (lane == 0) dst[row] = rsqrtf(fmaxf(s, 1e-12f));
}

// ---------------- fp32 WMMA GEMM: Sij and K = exp(5*clip(S)) ----------------
__global__ __launch_bounds__(256) void gemm_exp_kernel(const float* __restrict__ fA,
                                                       const float* __restrict__ fB,
                                                       const float* __restrict__ invA,
                                                       const float* __restrict__ invB,
                                                       float* __restrict__ outS,
                                                       float* __restrict__ Kmat) {
    int wave = threadIdx.x >> 5;
    int lane = threadIdx.x & 31;
    int tile = blockIdx.x * 8 + wave;       // 65536 16x16 tiles
    int b  = tile >> 12;
    int ti = (tile >> 6) & 63;
    int tj = tile & 63;
    int half = lane >> 4;
    int l16  = lane & 15;

    const float* Arow = fA + (size_t)(b * NN + ti * 16 + l16) * DD;
    const float* Brow = fB + (size_t)(b * NN + tj * 16 + l16) * DD;

    v8f acc = {};
    #pragma unroll 4
    for (int k = 0; k < DD; k += 4) {
        // A 16x4 f32 fragment: VGPR0 = K{0,2}, VGPR1 = K{1,3} -> contiguous pair per lane
        v2f a  = *(const v2f*)(Arow + k + 2 * half);
        v2f bb = *(const v2f*)(Brow + k + 2 * half);
        acc = __builtin_amdgcn_wmma_f32_16x16x4_f32(false, a, false, bb,
                                                    (short)0, acc, false, false);
    }

    float sb = invB[b * NN + tj * 16 + l16];
    size_t bbase = (size_t)b << 20;   // b * N * N
    #pragma unroll
    for (int r = 0; r < 8; ++r) {
        int row = ti * 16 + r + 8 * half;
        float sa = invA[b * NN + row];
        float s  = acc[r] * sa * sb;                       // Sij (normalized dot)
        size_t idx = bbase + (size_t)row * NN + tj * 16 + l16;
        outS[idx] = s;
        float cl = fminf(fmaxf(s, -3.0f), 3.0f);           // exp(-LAM*Cij) = exp(5*clip(S))
        Kmat[idx] = expf(5.0f * cl);
    }
}

// ---------------- grid-wide barrier ----------------
__device__ __forceinline__ void grid_barrier(unsigned* bar) {
    __syncthreads();
    if (threadIdx.x == 0) {
        unsigned g = __hip_atomic_load(&bar[1], __ATOMIC_RELAXED, __HIP_MEMORY_SCOPE_AGENT);
        unsigned a = __hip_atomic_fetch_add(&bar[0], 1u, __ATOMIC_ACQ_REL, __HIP_MEMORY_SCOPE_AGENT);
        if (a == NBLK - 1) {
            __hip_atomic_store(&bar[0], 0u, __ATOMIC_RELAXED, __HIP_MEMORY_SCOPE_AGENT);
            __hip_atomic_fetch_add(&bar[1], 1u, __ATOMIC_RELEASE, __HIP_MEMORY_SCOPE_AGENT);
        } else {
            while (__hip_atomic_load(&bar[1], __ATOMIC_ACQUIRE, __HIP_MEMORY_SCOPE_AGENT) == g)
                __builtin_amdgcn_s_sleep(8);
        }
    }
    __syncthreads();
}

// ---------------- persistent Sinkhorn kernel ----------------
// One block per 64-row band; band of K held in LDS (256 KB via TDM), 100 iters,
// final straight-through step + P_out write.
__global__ __launch_bounds__(256, 1) void sinkhorn_kernel(const float* __restrict__ Kmat,
                                                          float* __restrict__ cbuf,
                                                          float* __restrict__ Pout,
                                                          unsigned* __restrict__ bar) {
    __shared__ float band[BAND * NN];   // 256 KB (<= 320 KB/WGP on CDNA5)
    __shared__ float v_lds[NN];
    __shared__ float u_lds[BAND];

    const int tid   = threadIdx.x;
    const int wave  = tid >> 5;
    const int lane  = tid & 31;
    const int b     = blockIdx.x >> 4;
    const int chunk = blockIdx.x & 15;
    const int i0    = chunk * BAND;

    // ---- TDM: DMA the 64x1024 f32 band of K into LDS (issued by wave 0 only) ----
    if (tid < 32) {
        unsigned lds_off = (unsigned)(size_t)(&band[0]);
        unsigned long long ga =
            (unsigned long long)(size_t)(Kmat + (size_t)(b * NN + i0) * NN);
        v4u g0;
        g0[0] = 1u;                                   // count=1 valid descriptor
        g0[1] = lds_off;                              // LDS byte address
        g0[2] = (unsigned)ga;                         // global_addr[31:0]
        g0[3] = (unsigned)((ga >> 32) & 0x01FFFFFFu)  // global_addr[56:32]
              | 0x80000000u;                          // type=2 ("image")
        v8i g1;
        g1[0] = 0x00020000;        // workgroup_mask=0, data_size=2 (4B)
        g1[1] = 0x04000000;        // tensor_dim0 = 1024 (low16 in bits 63:48)
        g1[2] = (BAND << 16);      // tensor_dim1 = 64 (low16 in bits 95:80)
        g1[3] = 0x04000000;        // tile_dim0 = 1024 (bits 127:112)
        g1[4] = BAND;              // tile_dim1 = 64, tile_dim2 = 0
        g1[5] = NN;                // tensor_dim0_stride = 1024
        g1[6] = 0;
        g1[7] = 0;
        v4i z4 = {0, 0, 0, 0};
#if __clang_major__ >= 23
        v8i z8 = {0, 0, 0, 0, 0, 0, 0, 0};
        __builtin_amdgcn_tensor_load_to_lds(g0, g1, z4, z4, z8, 0);
#else
        __builtin_amdgcn_tensor_load_to_lds(g0, g1, z4, z4, 0);
#endif
        __builtin_amdgcn_s_wait_tensorcnt(0);
    }
    __syncthreads();

    const float4* band4 = (const float4*)band;
    const float4* v4p   = (const float4*)v_lds;

    // triple-buffered column sums: W(t)=t%3 accumulate, R(t)=(t+2)%3 read, Z(t)=(t+1)%3 zero
    for (int t = 0; t <= NITER; ++t) {
        const int W = t % 3, R = (t + 2) % 3, Z = (t + 1) % 3;

        // v_j = 1 (t==0) or 1/colsum_j from previous iteration
        if (t == 0) {
            #pragma unroll
            for (int q = 0; q < 4; ++q) v_lds[tid * 4 + q] = 1.0f;
        } else {
            const float* cr = cbuf + R * (BATCH * NN) + b * NN;
            #pragma unroll
            for (int q = 0; q < 4; ++q)
                v_lds[tid * 4 + q] = 1.0f / ld_dev(cr + tid * 4 + q);
        }
        __syncthreads();

        // zero the buffer that iteration t+1 will accumulate into
        if (t < NITER && tid < BAND)
            cbuf[Z * (BATCH * NN) + b * NN + i0 + tid] = 0.0f;

        // u-phase: u_i = 1 / sum_j K_ij v_j   (8 rows per wave, LDS only)
        #pragma unroll
        for (int rr = 0; rr < 8; ++rr) {
            int i = wave * 8 + rr;
            float acc = 0.0f;
            #pragma unroll
            for (int jj = 0; jj < 8; ++jj) {
                float4 kk = band4[i * 256 + jj * 32 + lane];
                float4 vv = v4p[jj * 32 + lane];
                acc += kk.x * vv.x + kk.y * vv.y + kk.z * vv.z + kk.w * vv.w;
            }
            #pragma unroll
            for (int off = 16; off > 0; off >>= 1) acc += __shfl_xor(acc, off, 32);
            if (lane == 0) u_lds[i] = 1.0f / acc;
        }
        __syncthreads();

        // v-phase: partial column sums of u_i * K_ij over this band -> global atomics
        float4 a4 = {0.0f, 0.0f, 0.0f, 0.0f};
        #pragma unroll 8
        for (int i = 0; i < BAND; ++i) {
            float4 kk = band4[i * 256 + tid];
            float  ui = u_lds[i];
            a4.x += kk.x * ui; a4.y += kk.y * ui; a4.z += kk.z * ui; a4.w += kk.w * ui;
        }
        float* cw = cbuf + W * (BATCH * NN) + b * NN + tid * 4;
        atomicAdd(cw + 0, a4.x);
        atomicAdd(cw + 1, a4.y);
        atomicAdd(cw + 2, a4.z);
        atomicAdd(cw + 3, a4.w);

        grid_barrier(bar);
    }

    // After the loop: u_lds holds u' = 1/(K v100); buffer (NITER%3 == 1) holds s = K^T u'.
    // P_out[i,j] = u'_i * K_ij / s_j   (row-then-column normalization of final P)
    const float* sarr = cbuf + ((NITER) % 3) * (BATCH * NN) + b * NN;
    float4 si;
    si.x = 1.0f / ld_dev(sarr + tid * 4 + 0);
    si.y = 1.0f / ld_dev(sarr + tid * 4 + 1);
    si.z = 1.0f / ld_dev(sarr + tid * 4 + 2);
    si.w = 1.0f / ld_dev(sarr + tid * 4 + 3);

    float4* P4 = (float4*)Pout + (size_t)(b * NN + i0) * (NN / 4) + tid;
    #pragma unroll 4
    for (int i = 0; i < BAND; ++i) {
        float4 kk = band4[i * 256 + tid];
        float  u  = u_lds[i];
        float4 o;
        o.x = kk.x * u * si.x;
        o.y = kk.y * u * si.y;
        o.z = kk.z * u * si.z;
        o.w = kk.w * u * si.w;
        P4[(size_t)i * (NN / 4)] = o;
    }
}

// ---------------- launcher ----------------
extern "C" void kernel_launch(void* const* d_in, const int* in_sizes, int n_in,
                              void* d_out, int out_size, void* d_ws, size_t ws_size,
                              hipStream_t stream) {
    (void)in_sizes; (void)n_in; (void)out_size; (void)ws_size;
    const float* fA = (const float*)d_in[0];
    const float* fB = (const float*)d_in[1];

    float* out  = (float*)d_out;
    float* Pout = out;                                   // [16,1024,1024]
    float* Sout = out + (size_t)BATCH * NN * NN;         // [16,1024,1024]

    float*   ws   = (float*)d_ws;
    float*   invA = ws;                                  // 16384
    float*   invB = ws + BATCH * NN;                     // 16384
    float*   Kmat = ws + 2 * BATCH * NN;                 // 16M floats (64 MB)
    float*   cbuf = Kmat + (size_t)BATCH * NN * NN;      // 3*16*1024
    unsigned* bar = (unsigned*)(cbuf + 3 * BATCH * NN);  // 2

    init_kernel<<<(3 * BATCH * NN + 255) / 256, 256, 0, stream>>>(cbuf, bar);
    l2norm_kernel<<<(2 * BATCH * NN) / 8, 256, 0, stream>>>(fA, fB, invA, invB);
    gemm_exp_kernel<<<(BATCH * 64 * 64) / 8, 256, 0, stream>>>(fA, fB, invA, invB, Sout, Kmat);
    sinkhorn_kernel<<<NBLK, 256, 0, stream>>>(Kmat, cbuf, Pout, bar);
}